// Net_34729105555909
// MI455X (gfx1250) — compile-verified
//
#include <hip/hip_runtime.h>

typedef __attribute__((ext_vector_type(2))) float v2f;
typedef __attribute__((ext_vector_type(8))) float v8f;

// Gather one 3x3 tap for an active site. x is dense (B*784) f32, zero at
// inactive sites, so a plain bounds-checked read implements the submanifold
// rulebook gather exactly.
__device__ __forceinline__ float gtap(const float* __restrict__ x, int idx,
                                      int yy, int xx, int dy, int dx) {
    int ny = yy + dy;
    int nx = xx + dx;
    if ((unsigned)ny < 28u && (unsigned)nx < 28u)
        return x[idx + dy * 28 + dx];
    return 0.0f;
}

// One wave handles 16 active sites x 32 output channels:
//   D(16x32) = G(16x12 zero-padded) @ Wpad(12x32) + bias
// as two 16x16 f32 accumulators, 3x V_WMMA_F32_16X16X4_F32 each.
__global__ __launch_bounds__(256) void sparse_conv3x3_wmma(
    const float* __restrict__ x,       // (B*784) dense, zeros at inactive sites
    const float* __restrict__ weight,  // (3,3,1,32) -> flat (9,32)
    const float* __restrict__ bias,    // (32)
    const int*   __restrict__ aidx,    // (N) flat b*784 + y*28 + x
    float*       __restrict__ out,     // (N,32)
    int N)
{
    __shared__ float lw[12 * 32];  // K padded 9 -> 12 with zero rows
    __shared__ float lb[32];

    int tid = threadIdx.x;
    for (int i = tid; i < 12 * 32; i += 256)
        lw[i] = (i < 9 * 32) ? weight[i] : 0.0f;
    if (tid < 32) lb[tid] = bias[tid];
    __syncthreads();

    int lane = tid & 31;
    int wave = tid >> 5;
    int row  = lane & 15;   // M row (lanes 0-15) / also output column index
    int half = lane >> 4;   // which half-wave

    int tile = blockIdx.x * 8 + wave;  // 16 sites per tile
    int base = tile * 16;
    int s  = base + row;
    int sc = s < N ? s : N - 1;        // clamp: keep EXEC all-1s for WMMA

    int idx = aidx[sc];
    unsigned rem = (unsigned)idx % 784u;
    int yy = (int)(rem / 28u);
    int xx = (int)(rem % 28u);

    // ---- A operands: taps t = ky*3+kx, K-chunks {0..3},{4..7},{8..11} ----
    // A 16x4 layout: VGPR0 = K=2*half, VGPR1 = K=2*half+1 (per chunk).
    v2f a0, a1, a2;
    if (half == 0) {
        a0.x = gtap(x, idx, yy, xx, -1, -1);  // tap 0
        a0.y = gtap(x, idx, yy, xx, -1,  0);  // tap 1
        a1.x = gtap(x, idx, yy, xx,  0,  0);  // tap 4
        a1.y = gtap(x, idx, yy, xx,  0,  1);  // tap 5
        a2.x = gtap(x, idx, yy, xx,  1,  1);  // tap 8
        a2.y = 0.0f;                          // pad K=9
    } else {
        a0.x = gtap(x, idx, yy, xx, -1,  1);  // tap 2
        a0.y = gtap(x, idx, yy, xx,  0, -1);  // tap 3
        a1.x = gtap(x, idx, yy, xx,  1, -1);  // tap 6
        a1.y = gtap(x, idx, yy, xx,  1,  0);  // tap 7
        a2.x = 0.0f;                          // pad K=10
        a2.y = 0.0f;                          // pad K=11
    }

    // ---- B operands: B(4x16) VGPR0 holds rows K, K+2 across lane halves ----
    int kb = 2 * half;
    v2f b0c0, b0c1, b1c0, b1c1, b2c0, b2c1;
    b0c0.x = lw[(0 + kb) * 32 + row];      b0c0.y = lw[(1 + kb) * 32 + row];
    b0c1.x = lw[(0 + kb) * 32 + row + 16]; b0c1.y = lw[(1 + kb) * 32 + row + 16];
    b1c0.x = lw[(4 + kb) * 32 + row];      b1c0.y = lw[(5 + kb) * 32 + row];
    b1c1.x = lw[(4 + kb) * 32 + row + 16]; b1c1.y = lw[(5 + kb) * 32 + row + 16];
    b2c0.x = lw[(8 + kb) * 32 + row];      b2c0.y = lw[(9 + kb) * 32 + row];
    b2c1.x = lw[(8 + kb) * 32 + row + 16]; b2c1.y = lw[(9 + kb) * 32 + row + 16];

    // ---- C init: every row of the tile gets the same bias[col] ----
    float bz0 = lb[row];
    float bz1 = lb[row + 16];
    v8f c0 = {bz0, bz0, bz0, bz0, bz0, bz0, bz0, bz0};
    v8f c1 = {bz1, bz1, bz1, bz1, bz1, bz1, bz1, bz1};

    // ---- D = A x B + C, chained over 3 K-chunks, two column halves ----
    c0 = __builtin_amdgcn_wmma_f32_16x16x4_f32(false, a0, false, b0c0, (short)0, c0, false, false);
    c0 = __builtin_amdgcn_wmma_f32_16x16x4_f32(false, a1, false, b1c0, (short)0, c0, false, false);
    c0 = __builtin_amdgcn_wmma_f32_16x16x4_f32(false, a2, false, b2c0, (short)0, c0, false, false);
    c1 = __builtin_amdgcn_wmma_f32_16x16x4_f32(false, a0, false, b0c1, (short)0, c1, false, false);
    c1 = __builtin_amdgcn_wmma_f32_16x16x4_f32(false, a1, false, b1c1, (short)0, c1, false, false);
    c1 = __builtin_amdgcn_wmma_f32_16x16x4_f32(false, a2, false, b2c1, (short)0, c1, false, false);

    // ---- Store ----
    // D layout: VGPR v holds rows M=v (lanes 0-15) and M=v+8 (lanes 16-31).
    // One base pointer; all 16 stores use compile-time immediate offsets.
    // Non-temporal: output is streamed once, keep it out of L2 so the 51 MB
    // x array stays resident for the 9-way gather.
    int site0 = base + 8 * half;
    float* p = out + (size_t)site0 * 32 + row;

    if (base + 16 <= N) {
        // Fast path: full tile in-bounds (always taken when N % 16 == 0).
#pragma unroll
        for (int v = 0; v < 8; ++v) {
            __builtin_nontemporal_store(c0[v], p + 32 * v);
            __builtin_nontemporal_store(c1[v], p + 32 * v + 16);
        }
    } else {
        // Cold tail: guard per row.
#pragma unroll
        for (int v = 0; v < 8; ++v) {
            if (site0 + v < N) {
                __builtin_nontemporal_store(c0[v], p + 32 * v);
                __builtin_nontemporal_store(c1[v], p + 32 * v + 16);
            }
        }
    }
}

extern "C" void kernel_launch(void* const* d_in, const int* in_sizes, int n_in,
                              void* d_out, int out_size, void* d_ws, size_t ws_size,
                              hipStream_t stream) {
    const float* x      = (const float*)d_in[0];  // (16384,28,28,1)
    const float* weight = (const float*)d_in[1];  // (3,3,1,32)
    const float* bias   = (const float*)d_in[2];  // (32)
    const int*   aidx   = (const int*)d_in[3];    // (N)
    float*       out    = (float*)d_out;          // (N,32)

    int N = in_sizes[3];
    int tiles  = (N + 15) / 16;        // 16 sites per wave
    int blocks = (tiles + 7) / 8;      // 8 waves per 256-thread block

    sparse_conv3x3_wmma<<<blocks, 256, 0, stream>>>(x, weight, bias, aidx, out, N);
}